// LSTMA_31361851195434
// MI455X (gfx1250) — compile-verified
//
#include <hip/hip_runtime.h>
#include <hip/hip_bf16.h>
#include <math.h>

// ---------------- constants (from reference) ----------------
#define DIM   512      // D
#define HID   512      // H
#define G3H   1536     // 3*H
#define TOPK  64
#define OUTN  5
#define PREDN 1537     // D + 2H + 1

typedef __attribute__((ext_vector_type(2))) float v2f;
typedef __attribute__((ext_vector_type(8))) float v8f;

// ---------------- shared: in-LDS bitonic sort (descending) ----------------
__device__ __forceinline__ void bitonic_desc(float* v, int* ix, int n, int tid, int nthr) {
    for (int k = 2; k <= n; k <<= 1) {
        for (int j = k >> 1; j > 0; j >>= 1) {
            __syncthreads();
            for (int i = tid; i < n; i += nthr) {
                int p = i ^ j;
                if (p > i) {
                    bool up = ((i & k) == 0);
                    float a = v[i], b = v[p];
                    bool sw = up ? (a < b) : (a > b);   // descending overall
                    if (sw) {
                        v[i] = b; v[p] = a;
                        int t = ix[i]; ix[i] = ix[p]; ix[p] = t;
                    }
                }
            }
        }
    }
    __syncthreads();
}

// ---------------- K1: alpha = fs @ fea (HBM-bound streaming mat-vec) ------
// One row per wave: 32 lanes x float4 = 512B coalesced per step, 4 steps/row,
// then cross-lane reduction. fea staged into LDS via CDNA5 async load-to-LDS.
__global__ __launch_bounds__(256) void k_alpha(const float* __restrict__ fs,
                                               const float* __restrict__ fea,
                                               float* __restrict__ alpha, int T) {
    __shared__ __align__(16) float sfea[DIM];
    const int tid = threadIdx.x;
    if (tid < 128) {
        // per-lane: copies 16 bytes from global memory into LDS  (ASYNCcnt tracked)
        unsigned ldsa = (unsigned)(uintptr_t)(&sfea[tid * 4]); // low 32b of flat = LDS offset
        const float* gp = fea + tid * 4;
        asm volatile("global_load_async_to_lds_b128 %0, %1, off"
                     :: "v"(ldsa), "v"(gp) : "memory");
    }
    asm volatile("s_wait_asynccnt 0" ::: "memory");
    __syncthreads();

    const int lane = tid & 31;
    const int gw   = blockIdx.x * (blockDim.x >> 5) + (tid >> 5);
    const int nw   = gridDim.x * (blockDim.x >> 5);

    // hoist the lane's slice of fea (invariant across rows)
    float4 b0 = *(const float4*)&sfea[(0 * 32 + lane) * 4];
    float4 b1 = *(const float4*)&sfea[(1 * 32 + lane) * 4];
    float4 b2 = *(const float4*)&sfea[(2 * 32 + lane) * 4];
    float4 b3 = *(const float4*)&sfea[(3 * 32 + lane) * 4];

    for (int r = gw; r < T; r += nw) {
        const float4* row = (const float4*)(fs + (size_t)r * DIM);
        float4 a0 = row[0 * 32 + lane];
        float4 a1 = row[1 * 32 + lane];
        float4 a2 = row[2 * 32 + lane];
        float4 a3 = row[3 * 32 + lane];
        float p = 0.f;
        p = fmaf(a0.x, b0.x, p); p = fmaf(a0.y, b0.y, p); p = fmaf(a0.z, b0.z, p); p = fmaf(a0.w, b0.w, p);
        p = fmaf(a1.x, b1.x, p); p = fmaf(a1.y, b1.y, p); p = fmaf(a1.z, b1.z, p); p = fmaf(a1.w, b1.w, p);
        p = fmaf(a2.x, b2.x, p); p = fmaf(a2.y, b2.y, p); p = fmaf(a2.z, b2.z, p); p = fmaf(a2.w, b2.w, p);
        p = fmaf(a3.x, b3.x, p); p = fmaf(a3.y, b3.y, p); p = fmaf(a3.z, b3.z, p); p = fmaf(a3.w, b3.w, p);
        #pragma unroll
        for (int m = 16; m > 0; m >>= 1) p += __shfl_xor(p, m, 32);
        if (lane == 0) alpha[r] = p;
    }
}

// ---------------- K2: hierarchical top-64 via bitonic sort ----------------
// Each block sorts up to 4096 (val,idx) pairs in LDS, emits its top 64.
__global__ __launch_bounds__(1024) void k_topk_reduce(const float* __restrict__ in_v,
                                                      const int* __restrict__ in_i,
                                                      int N,
                                                      float* __restrict__ out_v,
                                                      int* __restrict__ out_i) {
    __shared__ float sv[4096];
    __shared__ int   si[4096];
    const int tid  = threadIdx.x;
    const int base = blockIdx.x * 4096;
    for (int t = tid; t < 4096; t += 1024) {
        int g = base + t;
        if (g < N) { sv[t] = in_v[g]; si[t] = in_i ? in_i[g] : g; }
        else       { sv[t] = -INFINITY; si[t] = 0; }
    }
    bitonic_desc(sv, si, 4096, tid, 1024);
    if (tid < TOPK) {
        out_v[blockIdx.x * TOPK + tid] = sv[tid];
        out_i[blockIdx.x * TOPK + tid] = si[tid];
    }
}

// ---------------- K3: softmax over top-64 values --------------------------
__global__ void k_softmax64(const float* __restrict__ v, float* __restrict__ w) {
    __shared__ float se[TOPK];
    __shared__ float ssum;
    int t = threadIdx.x;
    se[t] = expf(v[t] - v[0]);   // sorted descending -> v[0] is max
    __syncthreads();
    if (t == 0) { float s = 0.f; for (int k = 0; k < TOPK; ++k) s += se[k]; ssum = s; }
    __syncthreads();
    w[t] = se[t] / ssum;
}

// ---------------- K4: attn_h = w @ hs[idx] (coalesced over j) -------------
__global__ void k_attn(const float* __restrict__ hs, const float* __restrict__ w,
                       const int* __restrict__ idx, float* __restrict__ attn) {
    int j = threadIdx.x;            // 512 threads
    float s = 0.f;
    for (int k = 0; k < TOPK; ++k)
        s = fmaf(w[k], hs[(size_t)idx[k] * HID + j], s);
    attn[j] = s;
}

// ---------------- K5: GRU GEMVs via V_WMMA_F32_16X16X4_F32 ----------------
// D = A x B + C with A[m][k] = x[c+k] broadcast over m, B[k][n] = W[o_base+n][c+k].
// Every row of D then holds the 16 dot products; lanes 0..15 of acc[0] = outputs.
// A 16x4 f32 frag (2 VGPRs): lanes 0-15 hold K=0,1 ; lanes 16-31 hold K=2,3.
// B 4x16 f32 frag mirrors: N = lane&15, K split across lane halves.
__global__ __launch_bounds__(256) void k_gru_wmma(const float* __restrict__ w_ih,
                                                  const float* __restrict__ w_hh,
                                                  const float* __restrict__ fea,
                                                  const float* __restrict__ hvec,
                                                  const float* __restrict__ b_ih,
                                                  const float* __restrict__ b_hh,
                                                  float* __restrict__ gi,
                                                  float* __restrict__ gh) {
    __shared__ float sx[2 * DIM];          // [0:512)=fea, [512:1024)=h
    const int tid = threadIdx.x;
    for (int i = tid; i < DIM; i += 256) { sx[i] = fea[i]; sx[DIM + i] = hvec[i]; }
    __syncthreads();

    const int gwave = blockIdx.x * 8 + (tid >> 5);   // 0..191 (24 blocks x 8 waves)
    const int lane  = tid & 31;
    const int mat   = (gwave >= 96);                 // 0: w_ih@fea -> gi ; 1: w_hh@h -> gh
    const int obase = (mat ? gwave - 96 : gwave) * 16;

    const float* W    = mat ? w_hh : w_ih;
    const float* x    = &sx[mat * DIM];
    const float* bias = mat ? b_hh : b_ih;
    float*       out  = mat ? gh : gi;

    const int n  = lane & 15;
    const int kk = (lane >> 4) << 1;                 // 0 or 2
    const float* wrow = W + (size_t)(obase + n) * DIM + kk;
    const float* xp   = x + kk;

    v8f acc = {0.f, 0.f, 0.f, 0.f, 0.f, 0.f, 0.f, 0.f};
    #pragma unroll 4
    for (int c = 0; c < DIM; c += 4) {
        v2f a; a.x = xp[c];   a.y = xp[c + 1];       // A frag: x broadcast over M
        v2f b; b.x = wrow[c]; b.y = wrow[c + 1];     // B frag: 16 weight rows along N
        acc = __builtin_amdgcn_wmma_f32_16x16x4_f32(false, a, false, b,
                                                    (short)0, acc, false, false);
    }
    if (lane < 16) out[obase + lane] = acc[0] + bias[obase + lane];
}

// ---------------- K6: GRU gates (torch order r,z,n) -> h_new --------------
__global__ void k_gates(const float* __restrict__ gi, const float* __restrict__ gh,
                        const float* __restrict__ hvec, float* __restrict__ d_out) {
    int j = threadIdx.x;            // 512 threads
    float r  = 1.f / (1.f + expf(-(gi[j]           + gh[j])));
    float z  = 1.f / (1.f + expf(-(gi[HID + j]     + gh[HID + j])));
    float nn = tanhf(gi[2 * HID + j] + r * gh[2 * HID + j]);
    d_out[OUTN + j] = (1.f - z) * nn + z * hvec[j];
}

// ---------------- K7: prediction head + log_softmax -----------------------
__global__ void k_head(const float* __restrict__ W_out, const float* __restrict__ b_out,
                       const float* __restrict__ fea, const float* __restrict__ attn,
                       const float* __restrict__ hvec, float* __restrict__ d_out) {
    __shared__ float sl[OUTN];
    const int wave = threadIdx.x >> 5, lane = threadIdx.x & 31;   // 256 threads
    if (wave < OUTN) {
        const float* wr = W_out + wave * PREDN;
        float s = 0.f;
        for (int e = lane; e < PREDN; e += 32) {
            float p = (e < DIM)       ? fea[e]
                    : (e < 2 * DIM)   ? attn[e - DIM]
                    : (e < 3 * DIM)   ? hvec[e - 2 * DIM]
                    : (float)TOPK;                       // length = K
            s = fmaf(wr[e], p, s);
        }
        #pragma unroll
        for (int m = 16; m > 0; m >>= 1) s += __shfl_xor(s, m, 32);
        if (lane == 0) sl[wave] = s + b_out[wave];
    }
    __syncthreads();
    if (threadIdx.x == 0) {
        float mx = sl[0];
        for (int o = 1; o < OUTN; ++o) mx = fmaxf(mx, sl[o]);
        float se = 0.f;
        for (int o = 0; o < OUTN; ++o) se += expf(sl[o] - mx);
        float lse = logf(se);
        for (int o = 0; o < OUTN; ++o) d_out[o] = sl[o] - mx - lse;
    }
}

// ---------------- host: orchestration -------------------------------------
extern "C" void kernel_launch(void* const* d_in, const int* in_sizes, int n_in,
                              void* d_out, int out_size, void* d_ws, size_t ws_size,
                              hipStream_t stream) {
    const float* fea   = (const float*)d_in[0];
    const float* h     = (const float*)d_in[1];
    const float* fs    = (const float*)d_in[2];
    const float* hs    = (const float*)d_in[3];
    const float* w_ih  = (const float*)d_in[4];
    const float* w_hh  = (const float*)d_in[5];
    const float* b_ih  = (const float*)d_in[6];
    const float* b_hh  = (const float*)d_in[7];
    const float* W_out = (const float*)d_in[8];
    const float* b_out = (const float*)d_in[9];
    const int T = in_sizes[2] / DIM;    // history length (200000)

    // workspace layout (floats)
    float* ws      = (float*)d_ws;
    float* alpha   = ws;                                  // T
    size_t off     = (size_t)((T + 255) & ~255);
    float* candAv  = ws + off;                            // <= 8192
    int*   candAi  = (int*)(ws + off + 8192);
    float* candBv  = ws + off + 16384;
    int*   candBi  = (int*)(ws + off + 24576);
    float* wsm     = ws + off + 32768;                    // 64 softmax weights
    float* attn    = wsm + TOPK;                          // 512
    float* gi      = attn + HID;                          // 1536
    float* gh      = gi + G3H;                            // 1536

    // 1) streaming mat-vec (the HBM-bound 410MB pass)
    k_alpha<<<1024, 256, 0, stream>>>(fs, fea, alpha, T);

    // 2) hierarchical top-64 select: 200000 -> 3136 -> 64
    const float* curv = alpha; const int* curi = nullptr; int N = T;
    float* ov = candAv; int* oi = candAi;
    float* sv2 = candBv; int* si2 = candBi;
    while (N > TOPK) {
        int nb = (N + 4095) / 4096;
        k_topk_reduce<<<nb, 1024, 0, stream>>>(curv, curi, N, ov, oi);
        curv = ov; curi = oi; N = nb * TOPK;
        float* tv = ov; ov = sv2; sv2 = tv;
        int* ti = oi; oi = si2; si2 = ti;
    }

    // 3) softmax over the 64 selected scores
    k_softmax64<<<1, TOPK, 0, stream>>>(curv, wsm);

    // 4) attention gather/blend
    k_attn<<<1, HID, 0, stream>>>(hs, wsm, curi, attn);

    // 5) GRU GEMVs on the WMMA path (192 waves, 16 outputs each)
    k_gru_wmma<<<24, 256, 0, stream>>>(w_ih, w_hh, fea, h, b_ih, b_hh, gi, gh);

    // 6) gates -> d_out[5..517)
    k_gates<<<1, HID, 0, stream>>>(gi, gh, h, (float*)d_out);

    // 7) head + log_softmax -> d_out[0..5)
    k_head<<<1, 256, 0, stream>>>(W_out, b_out, fea, attn, h, (float*)d_out);
}